// CausalMessagePassingLayer_59536836657879
// MI455X (gfx1250) — compile-verified
//
#include <hip/hip_runtime.h>

#define B_ 16
#define S_ 8192
#define D_ 256
#define E_ 4096
#define M_ 32768

typedef __attribute__((ext_vector_type(2))) float v2f;
typedef __attribute__((ext_vector_type(8))) float v8f;

// Native no-return fp32 global atomic add at device scope (L2 atomic units).
// gfx1250 uses GFX12-style VGLOBAL syntax; no-return form increments STOREcnt,
// covered by s_endpgm's implicit wait-idle.
__device__ __forceinline__ void atom_add_f32(float* p, float v) {
  asm volatile("global_atomic_add_f32 %0, %1, off scope:SCOPE_DEV"
               :: "v"(p), "v"(v) : "memory");
}

// ---------------------------------------------------------------------------
// 1) degree init: deg[b][e] = 1 (self loop)
__global__ void k_deginit(unsigned* __restrict__ deg) {
  int i = blockIdx.x * blockDim.x + threadIdx.x;
  if (i < B_ * E_) deg[i] = 1u;
}

// 2) degree count over col = ei[1]
__global__ void k_degcnt(const int* __restrict__ ei, unsigned* __restrict__ deg) {
  int i = blockIdx.x * blockDim.x + threadIdx.x;
  if (i >= B_ * M_) return;
  int b = i >> 15;           // M_ = 2^15
  int m = i & (M_ - 1);
  int col = ei[(size_t)b * 2 * M_ + M_ + m];
  atomicAdd(&deg[b * E_ + col], 1u);
}

// 3) xw[b][r][n] = sum_k temb[b][t2e[b][r]][k] * W[n][k]   (x @ W^T)
//    One wave -> 16 rows x 64 cols tile via V_WMMA_F32_16X16X4_F32.
__global__ void k_gemm(const float* __restrict__ temb, const int* __restrict__ t2e,
                       const float* __restrict__ Wm, float* __restrict__ xw) {
  const int gid  = blockIdx.x * blockDim.x + threadIdx.x;
  const int wave = gid >> 5;
  const int lane = threadIdx.x & 31;
  if (wave >= B_ * (E_ / 16) * (D_ / 64)) return;

  const int b    = wave >> 10;        // 1024 waves per sample
  const int w    = wave & 1023;
  const int rt   = w >> 2;            // row tile 0..255
  const int nb   = (w & 3) << 6;      // n base: 0,64,128,192
  const int half = lane >> 4;         // lane half selects K pair
  const int mr   = lane & 15;         // M (for A) / N (for B) within tile

  const int   row = (rt << 4) + mr;
  const int   src = t2e[b * E_ + row];
  const float* ap  = temb + ((size_t)b * S_ + src) * D_ + 2 * half;
  const float* wp0 = Wm + (size_t)(nb +  0 + mr) * D_ + 2 * half;
  const float* wp1 = Wm + (size_t)(nb + 16 + mr) * D_ + 2 * half;
  const float* wp2 = Wm + (size_t)(nb + 32 + mr) * D_ + 2 * half;
  const float* wp3 = Wm + (size_t)(nb + 48 + mr) * D_ + 2 * half;

  v8f acc[4] = {};

#if __has_builtin(__builtin_amdgcn_wmma_f32_16x16x4_f32)
  #pragma unroll 8
  for (int k = 0; k < D_; k += 4) {
    v2f av = *(const v2f*)(ap + k);
    v2f b0 = *(const v2f*)(wp0 + k);
    v2f b1 = *(const v2f*)(wp1 + k);
    v2f b2 = *(const v2f*)(wp2 + k);
    v2f b3 = *(const v2f*)(wp3 + k);
    acc[0] = __builtin_amdgcn_wmma_f32_16x16x4_f32(false, av, false, b0, (short)0, acc[0], false, false);
    acc[1] = __builtin_amdgcn_wmma_f32_16x16x4_f32(false, av, false, b1, (short)0, acc[1], false, false);
    acc[2] = __builtin_amdgcn_wmma_f32_16x16x4_f32(false, av, false, b2, (short)0, acc[2], false, false);
    acc[3] = __builtin_amdgcn_wmma_f32_16x16x4_f32(false, av, false, b3, (short)0, acc[3], false, false);
  }
#else
  // scalar fallback (should never be taken on gfx1250)
  for (int j = 0; j < 4; ++j)
    for (int r = 0; r < 8; ++r) {
      int rowm = (rt << 4) + 8 * half + r;
      int srcm = t2e[b * E_ + rowm];
      const float* am = temb + ((size_t)b * S_ + srcm) * D_;
      const float* wn = Wm + (size_t)(nb + j * 16 + mr) * D_;
      float s = 0.f;
      for (int k = 0; k < D_; ++k) s += am[k] * wn[k];
      acc[j][r] = s;
    }
#endif

  // D layout: element r of lane -> C[m = 8*half + r][n = nb + j*16 + mr]
  float* op = xw + ((size_t)b * E_ + (rt << 4) + 8 * half) * D_ + nb + mr;
  #pragma unroll
  for (int r = 0; r < 8; ++r) {
    op[(size_t)r * D_ +  0] = acc[0][r];
    op[(size_t)r * D_ + 16] = acc[1][r];
    op[(size_t)r * D_ + 32] = acc[2][r];
    op[(size_t)r * D_ + 48] = acc[3][r];
  }
}

// 4) gcn[b][e][d] = xw[b][e][d] / deg[b][e] + bias[d]   (self-loop message, norm = dinv^2)
__global__ void k_selfinit(const float* __restrict__ xw, const unsigned* __restrict__ deg,
                           const float* __restrict__ bias, float* __restrict__ gcn) {
  int i = blockIdx.x * blockDim.x + threadIdx.x;  // over B*E*(D/4)
  if (i >= B_ * E_ * (D_ / 4)) return;
  int d4 = (i & 63) << 2;
  int e  = (i >> 6) & (E_ - 1);
  int b  = i >> 18;                                // E_*64 = 2^18
  float di = 1.0f / (float)deg[b * E_ + e];
  size_t off = ((size_t)(b * E_ + e)) * D_ + d4;
  float4 x  = *(const float4*)(xw + off);
  float4 bb = *(const float4*)(bias + d4);
  float4 o;
  o.x = x.x * di + bb.x;  o.y = x.y * di + bb.y;
  o.z = x.z * di + bb.z;  o.w = x.w * di + bb.w;
  *(float4*)(gcn + off) = o;
}

// 5) gcn[col] += xw[row] * rsqrt(deg[row]) * rsqrt(deg[col]); one wave per edge
__global__ void k_edges(const float* __restrict__ xw, const unsigned* __restrict__ deg,
                        const int* __restrict__ ei, float* __restrict__ gcn) {
  int gid  = blockIdx.x * blockDim.x + threadIdx.x;
  int wv   = gid >> 5;
  int lane = threadIdx.x & 31;
  if (wv >= B_ * M_) return;
  int b = wv >> 15;
  int m = wv & (M_ - 1);
  const int* eb = ei + (size_t)b * 2 * M_;
  int row = eb[m];
  int col = eb[M_ + m];
  float nrm = rsqrtf((float)deg[b * E_ + row]) * rsqrtf((float)deg[b * E_ + col]);
  const float* src = xw  + ((size_t)(b * E_ + row)) * D_ + lane * 8;
  float*       dst = gcn + ((size_t)(b * E_ + col)) * D_ + lane * 8;
  float4 v0 = *(const float4*)(src);
  float4 v1 = *(const float4*)(src + 4);
  atom_add_f32(dst + 0, v0.x * nrm);
  atom_add_f32(dst + 1, v0.y * nrm);
  atom_add_f32(dst + 2, v0.z * nrm);
  atom_add_f32(dst + 3, v0.w * nrm);
  atom_add_f32(dst + 4, v1.x * nrm);
  atom_add_f32(dst + 5, v1.y * nrm);
  atom_add_f32(dst + 6, v1.z * nrm);
  atom_add_f32(dst + 7, v1.w * nrm);
}

// 6) residual copy: out = temb
__global__ void k_copy(const float4* __restrict__ in, float4* __restrict__ out, int n) {
  int i = blockIdx.x * blockDim.x + threadIdx.x;
  if (i < n) out[i] = in[i];
}

// 7) out[b][e2t[e]][:] = temb[b][e2t[e]][:] + (e ? gcn[b][e-1][:] : 0); one wave per (b,e)
__global__ void k_final(const float* __restrict__ temb, const int* __restrict__ e2t,
                        const float* __restrict__ gcn, float* __restrict__ out) {
  int gid  = blockIdx.x * blockDim.x + threadIdx.x;
  int wv   = gid >> 5;
  int lane = threadIdx.x & 31;
  if (wv >= B_ * E_) return;
  int b = wv >> 12;                  // E_ = 2^12
  int e = wv & (E_ - 1);
  int tok = e2t[b * E_ + e];
  size_t tb = ((size_t)b * S_ + tok) * D_ + lane * 8;
  float4 t0 = *(const float4*)(temb + tb);
  float4 t1 = *(const float4*)(temb + tb + 4);
  if (e > 0) {
    size_t gb = ((size_t)(b * E_ + e - 1)) * D_ + lane * 8;
    float4 c0 = *(const float4*)(gcn + gb);
    float4 c1 = *(const float4*)(gcn + gb + 4);
    t0.x += c0.x; t0.y += c0.y; t0.z += c0.z; t0.w += c0.w;
    t1.x += c1.x; t1.y += c1.y; t1.z += c1.z; t1.w += c1.w;
  }
  *(float4*)(out + tb)     = t0;
  *(float4*)(out + tb + 4) = t1;
}

extern "C" void kernel_launch(void* const* d_in, const int* in_sizes, int n_in,
                              void* d_out, int out_size, void* d_ws, size_t ws_size,
                              hipStream_t stream) {
  const float* temb = (const float*)d_in[0];
  const int*   t2e  = (const int*)d_in[1];
  const int*   ei   = (const int*)d_in[2];
  const int*   e2t  = (const int*)d_in[3];
  const float* Wm   = (const float*)d_in[4];
  const float* bias = (const float*)d_in[5];
  float*       out  = (float*)d_out;

  // workspace: [deg: B*E u32][xw: B*E*D f32][gcn: B*E*D f32]  (~134.5 MB)
  unsigned* deg = (unsigned*)d_ws;
  float*    xw  = (float*)((char*)d_ws + (size_t)B_ * E_ * sizeof(unsigned));
  float*    gcn = xw + (size_t)B_ * E_ * D_;

  k_deginit <<<(B_ * E_ + 255) / 256, 256, 0, stream>>>(deg);
  k_degcnt  <<<(B_ * M_ + 255) / 256, 256, 0, stream>>>(ei, deg);
  k_gemm    <<<(B_ * (E_ / 16) * (D_ / 64) * 32 + 255) / 256, 256, 0, stream>>>(temb, t2e, Wm, xw);
  k_selfinit<<<(B_ * E_ * (D_ / 4) + 255) / 256, 256, 0, stream>>>(xw, deg, bias, gcn);
  k_edges   <<<(B_ * M_ * 32 + 255) / 256, 256, 0, stream>>>(xw, deg, ei, gcn);
  k_copy    <<<(B_ * S_ * (D_ / 4) + 255) / 256, 256, 0, stream>>>((const float4*)temb, (float4*)out, B_ * S_ * (D_ / 4));
  k_final   <<<(B_ * E_ * 32 + 255) / 256, 256, 0, stream>>>(temb, e2t, gcn, out);
}